// PPM_25709674234322
// MI455X (gfx1250) — compile-verified
//
#include <hip/hip_runtime.h>

// ---------------------------------------------------------------------------
// XRF self-attenuation forward model for MI455X (gfx1250, wave32, WMMA).
// One workgroup (256 threads = 8 wave32) per minibatch row m.
//   Phase A: stage SA (32x1024) -> LDS via ASYNC global->LDS copies (no VGPR
//            round-trip, overlaps the scan); exclusive attenuation scan over
//            depth; build W8T[e][n] = att[n]*xp[e][n] in LDS (element-major).
//   Phase B: fl[l,m] via V_WMMA_F32_16X16X4_F32:  C(16x16) += A(16x4)*B(4x16)
//            A = SA tile rows (lines) from LDS, B = W columns (e<8; cols 8..15
//            zeroed via a per-lane multiplier -> one v_pk_mul, no predication).
//            fl[l] = C[l&15][l>>2] * 5000 * fl_unit[l].
// Memory-bound (~80 MB @ 23.3 TB/s ~ 3.5 us); WMMA reduction is free FLOPs.
// ---------------------------------------------------------------------------

#define N_ELEMENT 8
#define N_LINES   32
#define MINIBATCH 512
#define SAMPLE_N  1024

static constexpr float SCALE      = 0.01f / 1024.0f;          // SAMPLE_CM / SAMPLE_N
static constexpr float OUT_FACTOR = 100000.0f * 0.05f * 1.0f; // I0 * solid_angle * att

typedef float v2f __attribute__((ext_vector_type(2)));
typedef float v8f __attribute__((ext_vector_type(8)));

// LDS layout (floats).
// SA rows: stride 1028 (== 4 mod 64) -> the 16 rows of an A-tile map to 16
// distinct bank groups for ds_load_b64. 4112 B/row keeps 16B alignment.
// W8T rows: stride 1032 (== 8 mod 64) -> 8 element rows on distinct banks.
#define SA_STRIDE  1028
#define W8T_STRIDE 1032
#define SA_OFF     0
#define W8T_OFF    (32 * SA_STRIDE)            // 32896
#define CSC_OFF    (W8T_OFF + 8 * W8T_STRIDE)  // 41152 (8 waves x 256 accum)
#define WS_OFF     (CSC_OFF + 8 * 256)         // 43200 (scan scratch)
#define SMEM_FLOATS (WS_OFF + 32)              // 43232 -> 172,928 bytes

// Raw LDS byte offset of a generic pointer that points into shared memory.
__device__ __forceinline__ unsigned lds_offset_of(const void* p) {
    return (unsigned)(uintptr_t)(__attribute__((address_space(3))) const void*)p;
}

__global__ __launch_bounds__(256, 1)
void PPM_25709674234322_kernel(const float* __restrict__ xp,
                               const float* __restrict__ mu_probe,
                               const float* __restrict__ fl_unit,
                               const float* __restrict__ SA,
                               float* __restrict__ out)
{
    extern __shared__ float sm[];
    float* sa_lds = sm + SA_OFF;
    float* w8t    = sm + W8T_OFF;
    float* csc    = sm + CSC_OFF;
    float* wsc    = sm + WS_OFF;

    const int m    = blockIdx.x;
    const int t    = threadIdx.x;
    const int lane = t & 31;
    const int wid  = t >> 5;

    // ------- Stage SA tile (32 lines x 1024 voxels) -> LDS, ASYNC, no VGPRs --
    // 8192 float4 total; thread t owns float4 index t + 256*i.
    // GLOBAL_LOAD_ASYNC_TO_LDS_B128 (GV mode): VDST = LDS byte offset,
    // VADDR = 64-bit global byte address. Tracked by ASYNCcnt.
    {
        const float* sa_base = SA + (size_t)m * SAMPLE_N;
        #pragma unroll
        for (int i = 0; i < 32; ++i) {
            int idx4 = t + 256 * i;
            int row  = idx4 >> 8;            // 256 float4 per line
            int pos  = (idx4 & 255) << 2;    // float offset within line
            unsigned lds_b = lds_offset_of(sa_lds + row * SA_STRIDE + pos);
            unsigned long long gaddr = (unsigned long long)(uintptr_t)(
                sa_base + (size_t)row * ((size_t)MINIBATCH * SAMPLE_N) + pos);
            asm volatile("global_load_async_to_lds_b128 %0, %1, off"
                         :
                         : "v"(lds_b), "v"(gaddr)
                         : "memory");
        }
    }

    // ---------------- Phase A: xp load + attenuation scan --------------------
    float mu[N_ELEMENT];
    #pragma unroll
    for (int e = 0; e < N_ELEMENT; ++e) mu[e] = mu_probe[e];  // scalarized

    const int nb = t * 4;                     // this thread owns voxels nb..nb+3
    float4 x4[N_ELEMENT];
    #pragma unroll
    for (int e = 0; e < N_ELEMENT; ++e)
        x4[e] = *reinterpret_cast<const float4*>(
            xp + ((size_t)(e * MINIBATCH + m)) * SAMPLE_N + nb);

    float s0 = 0.f, s1 = 0.f, s2 = 0.f, s3 = 0.f;
    #pragma unroll
    for (int e = 0; e < N_ELEMENT; ++e) {
        s0 = fmaf(mu[e], x4[e].x, s0);
        s1 = fmaf(mu[e], x4[e].y, s1);
        s2 = fmaf(mu[e], x4[e].z, s2);
        s3 = fmaf(mu[e], x4[e].w, s3);
    }
    const float i1 = s0 + s1, i2 = i1 + s2, T = i2 + s3;

    // wave32 inclusive scan of per-thread totals
    float sc = T;
    #pragma unroll
    for (int d = 1; d < 32; d <<= 1) {
        float v = __shfl_up(sc, d, 32);
        if (lane >= d) sc += v;
    }
    if (lane == 31) wsc[wid] = sc;
    __syncthreads();
    if (t == 0) {
        float acc = 0.f;
        #pragma unroll
        for (int w = 0; w < 8; ++w) { float v = wsc[w]; wsc[8 + w] = acc; acc += v; }
        wsc[16] = acc;                                        // row grand total
    }
    __syncthreads();
    const float base = wsc[8 + wid] + (sc - T);               // exclusive prefix
    if (t == 0) out[N_LINES * MINIBATCH + m] = SCALE * wsc[16]; // transmission

    // attenuation at each owned voxel (exclusive cumsum, matches [:, :-1])
    float att0 = __expf(-SCALE * base);
    float att1 = __expf(-SCALE * (base + s0));
    float att2 = __expf(-SCALE * (base + i1));
    float att3 = __expf(-SCALE * (base + i2));

    // W8T[e][n] = att[n] * xp[e][n]; lane-consecutive float4 stores (no bank
    // conflicts: consecutive lanes hit consecutive 16B chunks of one row).
    #pragma unroll
    for (int e = 0; e < N_ELEMENT; ++e) {
        float4 w;
        w.x = att0 * x4[e].x;
        w.y = att1 * x4[e].y;
        w.z = att2 * x4[e].z;
        w.w = att3 * x4[e].w;
        *reinterpret_cast<float4*>(w8t + e * W8T_STRIDE + nb) = w;
    }

    // All async SA copies must have landed in LDS before Phase B reads them.
    asm volatile("s_wait_asynccnt 0" ::: "memory");
    __syncthreads();   // SA tile + W8T visible to all waves

    // ---------------- Phase B: WMMA reduction --------------------------------
    // wave wid: M-tile = wid>>2 (lines 0-15 / 16-31), K-chunk = (wid&3)*256.
    // A 16x4 f32 layout: lane<16 -> M=lane, {K=0,K=1}; lane>=16 -> {K=2,K=3}.
    // B 4x16 f32 layout: VGPR r -> row K=r (lanes 0-15) / K=r+2 (lanes 16-31),
    //   column = lane&15. b.x/b.y are consecutive n -> one ds_load_b64.
    const int tile = wid >> 2;
    const int kc   = wid & 3;
    const int l    = lane & 15;
    const int hi16 = lane >> 4;
    const int k0   = kc * 256;

    const float* arow  = sa_lds + (tile * 16 + l) * SA_STRIDE + k0 + 2 * hi16;
    const float* brow  = w8t + (l & 7) * W8T_STRIDE + k0 + 2 * hi16;
    const float  bmask = (l < N_ELEMENT) ? 1.0f : 0.0f;  // zero cols 8..15

    v8f c = {};
    #pragma unroll 8
    for (int kk = 0; kk < 256; kk += 4) {                 // uniform trip count
        v2f a    = *reinterpret_cast<const v2f*>(arow + kk);  // ds_load_b64
        v2f braw = *reinterpret_cast<const v2f*>(brow + kk);  // ds_load_b64
        v2f b;
        b.x = braw.x * bmask;
        b.y = braw.y * bmask;
        c = __builtin_amdgcn_wmma_f32_16x16x4_f32(
                /*neg_a=*/false, a, /*neg_b=*/false, b,
                /*c_mod=*/(short)0, c, /*reuse_a=*/false, /*reuse_b=*/false);
    }

    // dump accumulators for cross-wave K reduction
    #pragma unroll
    for (int r = 0; r < 8; ++r) csc[wid * 256 + r * 32 + lane] = c[r];
    __syncthreads();

    // ---------------- Extract fl[l] = sum_kc C[l&15][l>>2] -------------------
    if (t < N_LINES) {
        const int li  = t;
        const int tl  = li >> 4;             // M-tile
        const int R   = li & 15;             // row within tile
        const int col = li >> 2;             // element index l/4 (B column)
        const int vg  = R & 7;               // C VGPR index
        const int ln  = col + 16 * (R >> 3); // lane holding (R, col)
        float s = 0.f;
        #pragma unroll
        for (int q = 0; q < 4; ++q)
            s += csc[(tl * 4 + q) * 256 + vg * 32 + ln];
        out[li * MINIBATCH + m] = OUT_FACTOR * fl_unit[li] * s;
    }
}

extern "C" void kernel_launch(void* const* d_in, const int* in_sizes, int n_in,
                              void* d_out, int out_size, void* d_ws, size_t ws_size,
                              hipStream_t stream) {
    (void)in_sizes; (void)n_in; (void)out_size; (void)d_ws; (void)ws_size;
    const float* xp       = (const float*)d_in[0];
    const float* mu_probe = (const float*)d_in[1];
    const float* fl_unit  = (const float*)d_in[2];
    const float* SA       = (const float*)d_in[3];
    float* out            = (float*)d_out;

    const size_t smem = (size_t)SMEM_FLOATS * sizeof(float); // 172,928 B < 320 KB
    PPM_25709674234322_kernel<<<MINIBATCH, 256, smem, stream>>>(
        xp, mu_probe, fl_unit, SA, out);
}